// GNN_55559696941085
// MI455X (gfx1250) — compile-verified
//
#include <hip/hip_runtime.h>

// ---------------------------------------------------------------------------
// GNN (5x GraphConv + global_add_pool + MLP) for MI455X / gfx1250, wave32.
// GEMMs use V_WMMA_F32_16X16X4_F32 (fp32 matrix pipe; workload is L2-bound,
// not compute-bound, so keeping reference fp32 precision is free).
// LDS staging uses GLOBAL_LOAD_ASYNC_TO_LDS_B128 (ASYNCcnt) when available.
// ---------------------------------------------------------------------------

typedef __attribute__((ext_vector_type(2))) float v2f;
typedef __attribute__((ext_vector_type(8))) float v8f;
typedef __attribute__((ext_vector_type(4))) int  int4v;

#if defined(__has_builtin)
#if __has_builtin(__builtin_amdgcn_global_load_async_to_lds_b128) && \
    __has_builtin(__builtin_amdgcn_s_wait_asynccnt)
#define USE_ASYNC_LDS 1
#endif
#endif
#ifndef USE_ASYNC_LDS
#define USE_ASYNC_LDS 0
#endif

#if USE_ASYNC_LDS
typedef __attribute__((address_space(1))) int4v* gptr_t;   // global (AS1)
typedef __attribute__((address_space(3))) int4v* lptr_t;   // LDS (AS3)
#endif

#define LDA 132   // 64-row A tile stride (132 % 64 = 4 -> conflict-free M reads)
#define LDW 132   // 128-row W tile stride

// Block: 256 threads = 8 waves. Block computes 64 rows x 128 cols of
//   Out = relu( A @ Wrel + bias [+ Hm @ Wroot] )
// Wave w owns cols [16w,16w+16) across 4 row tiles (B fragment reused 4x).
__global__ __launch_bounds__(256)
void gnn_gemm_wmma_kernel(const float* __restrict__ A,     // [N,128]
                          const float* __restrict__ Hm,    // [N,128]
                          const float* __restrict__ Wrel,  // [128,128]
                          const float* __restrict__ Wroot, // [128,128]
                          const float* __restrict__ bias,  // [128]
                          float* __restrict__ Out,         // [N,128]
                          int N, int nRowTiles, int hasRoot)
{
    __shared__ float sA[64 * LDA];
    __shared__ float sW[128 * LDW];

    const int tid  = threadIdx.x;
    const int wave = tid >> 5;
    const int lane = tid & 31;
    const int half = lane >> 4;   // K-pair selector per WMMA f32 layout
    const int l16  = lane & 15;   // M (A-frag) / N (B-frag, D) index
    const int colBase = wave * 16;
    const int rowBase = (int)blockIdx.x * 64;

    v8f acc[4] = {};

    for (int phase = 0; phase < 2; ++phase) {
        if (phase == 1 && !hasRoot) break;
        const float* Ain = (phase == 0) ? A : Hm;
        const float* Win = (phase == 0) ? Wrel : Wroot;

        __syncthreads();   // protect LDS from previous phase's readers

#if USE_ASYNC_LDS
        // Async DMA global->LDS (no VGPR round trip, single wait at end).
        #pragma unroll
        for (int j = 0; j < 8; ++j) {        // A tile: 2048 float4 / 256 thr
            int i  = tid + j * 256;
            int r  = i >> 5;
            int c4 = i & 31;
            int grow = rowBase + r; if (grow >= N) grow = N - 1;
            __builtin_amdgcn_global_load_async_to_lds_b128(
                (gptr_t)(Ain + (size_t)grow * 128 + c4 * 4),
                (lptr_t)(&sA[r * LDA + c4 * 4]), 0, 0);
        }
        #pragma unroll
        for (int j = 0; j < 16; ++j) {       // W tile: 4096 float4 / 256 thr
            int i  = tid + j * 256;
            int r  = i >> 5;
            int c4 = i & 31;
            __builtin_amdgcn_global_load_async_to_lds_b128(
                (gptr_t)(Win + r * 128 + c4 * 4),
                (lptr_t)(&sW[r * LDW + c4 * 4]), 0, 0);
        }
        __builtin_amdgcn_s_wait_asynccnt(0);
#else
        // Fallback: register-batched copy (issue all loads, then all stores).
        {
            float4 tmp[8];
            #pragma unroll
            for (int j = 0; j < 8; ++j) {
                int i = tid + j * 256;
                int r = i >> 5, c4 = i & 31;
                int grow = rowBase + r; if (grow >= N) grow = N - 1;
                tmp[j] = *(const float4*)(Ain + (size_t)grow * 128 + c4 * 4);
            }
            #pragma unroll
            for (int j = 0; j < 8; ++j) {
                int i = tid + j * 256;
                int r = i >> 5, c4 = i & 31;
                *(float4*)(&sA[r * LDA + c4 * 4]) = tmp[j];
            }
            #pragma unroll
            for (int b = 0; b < 2; ++b) {
                #pragma unroll
                for (int j = 0; j < 8; ++j) {
                    int i = tid + (b * 8 + j) * 256;
                    int r = i >> 5, c4 = i & 31;
                    tmp[j] = *(const float4*)(Win + r * 128 + c4 * 4);
                }
                #pragma unroll
                for (int j = 0; j < 8; ++j) {
                    int i = tid + (b * 8 + j) * 256;
                    int r = i >> 5, c4 = i & 31;
                    *(float4*)(&sW[r * LDW + c4 * 4]) = tmp[j];
                }
            }
        }
#endif
        __syncthreads();

        #pragma unroll 4
        for (int k = 0; k < 128; k += 4) {
            // B fragment: 4x16 slab, lane half selects K pair {k,k+1}/{k+2,k+3}
            v2f b;
            b.x = sW[(k + 2 * half)     * LDW + colBase + l16];
            b.y = sW[(k + 2 * half + 1) * LDW + colBase + l16];
            #pragma unroll
            for (int rt = 0; rt < 4; ++rt) {
                v2f a;
                const int m = rt * 16 + l16;
                a.x = sA[m * LDA + k + 2 * half];
                a.y = sA[m * LDA + k + 2 * half + 1];
                acc[rt] = __builtin_amdgcn_wmma_f32_16x16x4_f32(
                    false, a, false, b, (short)0, acc[rt], false, false);
            }
        }
    }

    // Epilogue: bias + ReLU, D layout: VGPR r -> row (r + 8*half), col l16.
    const float bv = bias[colBase + l16];
    #pragma unroll
    for (int rt = 0; rt < 4; ++rt) {
        const int rowTile = (int)blockIdx.x * 4 + rt;
        if (rowTile >= nRowTiles) continue;
        #pragma unroll
        for (int r = 0; r < 8; ++r) {
            const int row = rowTile * 16 + r + 8 * half;
            float v = acc[rt][r] + bv;
            v = fmaxf(v, 0.0f);
            Out[(size_t)row * 128 + colBase + l16] = v;
        }
    }
}

// One wave32 per edge: lane covers 4 of 128 features (float4 gather +
// 4x global_atomic_add_f32 scatter; accumulator is L2-resident).
__global__ __launch_bounds__(256)
void gnn_scatter_edges_kernel(const float* __restrict__ h,
                              const int* __restrict__ src,
                              const int* __restrict__ dst,
                              float* __restrict__ agg, int E)
{
    const int e = (int)blockIdx.x * 8 + (threadIdx.x >> 5);
    if (e >= E) return;
    const int lane = threadIdx.x & 31;
    const int s = src[e];
    const int d = dst[e];
    const float4 v = *(const float4*)(h + (size_t)s * 128 + lane * 4);
    float* p = agg + (size_t)d * 128 + lane * 4;
    atomicAdd(p + 0, v.x);
    atomicAdd(p + 1, v.y);
    atomicAdd(p + 2, v.z);
    atomicAdd(p + 3, v.w);
}

// global_add_pool: one wave32 per node.
__global__ __launch_bounds__(256)
void gnn_pool_kernel(const float* __restrict__ h,
                     const int* __restrict__ batch,
                     float* __restrict__ g, int N)
{
    const int n = (int)blockIdx.x * 8 + (threadIdx.x >> 5);
    if (n >= N) return;
    const int lane = threadIdx.x & 31;
    const int b = batch[n];
    const float4 v = *(const float4*)(h + (size_t)n * 128 + lane * 4);
    float* p = g + (size_t)b * 128 + lane * 4;
    atomicAdd(p + 0, v.x);
    atomicAdd(p + 1, v.y);
    atomicAdd(p + 2, v.z);
    atomicAdd(p + 3, v.w);
}

// fc2: out[g] = dot(h1[g,:], W) + b  (128 -> 1), one block per graph.
__global__ __launch_bounds__(128)
void gnn_fc2_kernel(const float* __restrict__ h1,
                    const float* __restrict__ W,
                    const float* __restrict__ b,
                    float* __restrict__ out, int H)
{
    __shared__ float red[128];
    const int g = blockIdx.x;
    const int j = threadIdx.x;
    red[j] = h1[(size_t)g * H + j] * W[j];
    __syncthreads();
    for (int s = 64; s > 0; s >>= 1) {
        if (j < s) red[j] += red[j + s];
        __syncthreads();
    }
    if (j == 0) out[g] = red[0] + b[0];
}

extern "C" void kernel_launch(void* const* d_in, const int* in_sizes, int n_in,
                              void* d_out, int out_size, void* d_ws, size_t ws_size,
                              hipStream_t stream)
{
    const float* x     = (const float*)d_in[0];
    const int*   ei    = (const int*)d_in[1];   // [2,E]: src row then dst row
    const int*   batch = (const int*)d_in[2];
    // d_in[3] = num_graphs scalar (G == out_size, use that)

    const float *Wr[5], *br[5], *Wl[5];
    for (int i = 0; i < 5; ++i) {
        Wr[i] = (const float*)d_in[4 + 3 * i];
        br[i] = (const float*)d_in[5 + 3 * i];
        Wl[i] = (const float*)d_in[6 + 3 * i];
    }
    const float* fc1W = (const float*)d_in[19];
    const float* fc1b = (const float*)d_in[20];
    const float* fc2W = (const float*)d_in[21];
    const float* fc2b = (const float*)d_in[22];

    const int H = in_sizes[5];          // 128 (b_rel1 length)
    const int N = in_sizes[0] / H;      // 50000
    const int E = in_sizes[1] / 2;      // 800000
    const int G = out_size;             // 256

    // Workspace layout (floats): h0 | h1 | agg | g | g2
    float* ws  = (float*)d_ws;
    float* h0  = ws;
    float* h1  = h0 + (size_t)N * H;
    float* agg = h1 + (size_t)N * H;
    float* g   = agg + (size_t)N * H;
    float* g2  = g + (size_t)G * H;
    (void)ws_size; (void)n_in;

    const int nRowTiles   = N / 16;                 // N is a multiple of 16
    const int gemmBlocks  = (nRowTiles + 3) / 4;
    const int scatterBlks = (E + 7) / 8;
    const int poolBlks    = (N + 7) / 8;

    const float* hc = x;
    float* bufs[2] = { h0, h1 };
    for (int l = 0; l < 5; ++l) {
        (void)hipMemsetAsync(agg, 0, (size_t)N * H * sizeof(float), stream);
        gnn_scatter_edges_kernel<<<scatterBlks, 256, 0, stream>>>(hc, ei, ei + E, agg, E);
        float* hn = bufs[l & 1];
        gnn_gemm_wmma_kernel<<<gemmBlocks, 256, 0, stream>>>(
            agg, hc, Wr[l], Wl[l], br[l], hn, N, nRowTiles, /*hasRoot=*/1);
        hc = hn;
    }

    (void)hipMemsetAsync(g, 0, (size_t)G * H * sizeof(float), stream);
    gnn_pool_kernel<<<poolBlks, 256, 0, stream>>>(hc, batch, g, N);

    const int gTiles = G / 16;
    gnn_gemm_wmma_kernel<<<(gTiles + 3) / 4, 256, 0, stream>>>(
        g, g, fc1W, fc1W, fc1b, g2, G, gTiles, /*hasRoot=*/0);

    gnn_fc2_kernel<<<G, 128, 0, stream>>>(g2, fc2W, fc2b, (float*)d_out, H);
}